// GHMC_8495445311491
// MI455X (gfx1250) — compile-verified
//
#include <hip/hip_runtime.h>
#include <math.h>

#define GHM_BINS 10
#define THREADS 256
#define NWAVES (THREADS / 32)

// ---------------------------------------------------------------------------
// feature probes (compile-only loop: degrade gracefully, inspect histogram)
// ---------------------------------------------------------------------------
#define HAS_ASYNC_LDS __has_builtin(__builtin_amdgcn_global_load_async_to_lds_b128)
#define HAS_WMMA_F32  __has_builtin(__builtin_amdgcn_wmma_f32_16x16x4_f32)

// ---------------------------------------------------------------------------
// wave32 ballot (gfx1250 is wave32-only)
// ---------------------------------------------------------------------------
__device__ __forceinline__ unsigned ballot32(bool p) {
#if __has_builtin(__builtin_amdgcn_ballot_w32)
  return __builtin_amdgcn_ballot_w32(p);
#else
  return (unsigned)__ballot(p);
#endif
}

// ---------------------------------------------------------------------------
// async-copy 16B global -> LDS (per lane) + split-counter waits
// builtin params are pointers to 16B int vectors (per round-2 diagnostic)
// ---------------------------------------------------------------------------
typedef int v4i_ __attribute__((ext_vector_type(4)));

#if HAS_ASYNC_LDS
__device__ __forceinline__ void async_cp_b128(const float4* g, float4* l) {
  __builtin_amdgcn_global_load_async_to_lds_b128(
      (__attribute__((address_space(1))) v4i_*)(g),
      (__attribute__((address_space(3))) v4i_*)(l), 0, 0);
}
#endif

__device__ __forceinline__ void wait_async0() {
#if __has_builtin(__builtin_amdgcn_s_wait_asynccnt)
  __builtin_amdgcn_s_wait_asynccnt(0);
#else
  asm volatile("s_wait_asynccnt 0x0" ::: "memory");
#endif
}
__device__ __forceinline__ void wait_async3() {
#if __has_builtin(__builtin_amdgcn_s_wait_asynccnt)
  __builtin_amdgcn_s_wait_asynccnt(3);
#else
  asm volatile("s_wait_asynccnt 0x3" ::: "memory");
#endif
}

// ---------------------------------------------------------------------------
// wave-wide f32 sum.  WMMA version: with A=(v,0) and B=ones(4x16),
// D1[m][*] = v[m]+v[m+16].  Four accumulating WMMAs with A=ones and B taken
// from consecutive VGPR pairs of D1 add rows {0,8,1,9},{2,10,3,11},
// {4,12,5,13},{6,14,7,15} -> every element of D2 is the full 32-lane sum.
// Requires EXEC=all-1s (call only from non-divergent code).
// ---------------------------------------------------------------------------
#if HAS_WMMA_F32
typedef float v2f __attribute__((ext_vector_type(2)));
typedef float v8f __attribute__((ext_vector_type(8)));

__device__ __forceinline__ float wave_sum32(float v) {
  v2f a;    a[0] = v;    a[1] = 0.0f;
  v2f one2; one2[0] = 1.0f; one2[1] = 1.0f;
  v8f z = {0.f, 0.f, 0.f, 0.f, 0.f, 0.f, 0.f, 0.f};
  v8f d1 = __builtin_amdgcn_wmma_f32_16x16x4_f32(false, a, false, one2,
                                                 (short)0, z, false, false);
  v8f d2 = z;
  v2f b;
  b[0] = d1[0]; b[1] = d1[1];
  d2 = __builtin_amdgcn_wmma_f32_16x16x4_f32(false, one2, false, b, (short)0, d2, false, false);
  b[0] = d1[2]; b[1] = d1[3];
  d2 = __builtin_amdgcn_wmma_f32_16x16x4_f32(false, one2, false, b, (short)0, d2, false, false);
  b[0] = d1[4]; b[1] = d1[5];
  d2 = __builtin_amdgcn_wmma_f32_16x16x4_f32(false, one2, false, b, (short)0, d2, false, false);
  b[0] = d1[6]; b[1] = d1[7];
  d2 = __builtin_amdgcn_wmma_f32_16x16x4_f32(false, one2, false, b, (short)0, d2, false, false);
  return d2[0];
}
#else
__device__ __forceinline__ float wave_sum32(float v) {
#pragma unroll
  for (int off = 16; off > 0; off >>= 1) v += __shfl_xor(v, off, 32);
  return v;
}
#endif

// ---------------------------------------------------------------------------
// element math
// ---------------------------------------------------------------------------
__device__ __forceinline__ int ghm_bin(float p, float t, float w) {
  if (!(w > 0.0f)) return -1;
  float s = 1.0f / (1.0f + expf(-p));  // sigmoid
  int b = (int)(fabsf(s - t) * (float)GHM_BINS);
  b = b < 0 ? 0 : b;
  return b > GHM_BINS - 1 ? GHM_BINS - 1 : b;
}

__device__ __forceinline__ float bce_logits(float p, float t) {
  return fmaxf(p, 0.0f) - p * t + log1pf(expf(-fabsf(p)));
}

// ---------------------------------------------------------------------------
// wave-uniform tile streamer over three float4 arrays.
// Async path: double-buffered GLOBAL_LOAD_ASYNC_TO_LDS_B128 pipeline (issue
// tile j+1, wait ASYNCcnt<=3 so tile j's 3 copies are done, consume from LDS).
// Fallback: direct b128 loads + global_prefetch_b8.
// fn(p,t,w,act) is invoked by the full wave every iteration (ballot-safe).
// ---------------------------------------------------------------------------
template <typename Fn>
__device__ __forceinline__ void stream_tiles(const float4* __restrict__ p4,
                                             const float4* __restrict__ t4,
                                             const float4* __restrict__ w4,
                                             long long total4,
                                             float4* __restrict__ bp,  // [2][32]
                                             float4* __restrict__ bt,
                                             float4* __restrict__ bw,
                                             Fn fn) {
  const int tid = threadIdx.x;
  const int lane = tid & 31;
  const long long stride = (long long)gridDim.x * blockDim.x;
  const long long base0 =
      (long long)blockIdx.x * blockDim.x + (long long)(tid & ~31);

#if HAS_ASYNC_LDS
  int cur = 0;
  bool curFull = (base0 < total4) && (base0 + 32 <= total4);
  if (curFull) {
    async_cp_b128(p4 + base0 + lane, bp + lane);
    async_cp_b128(t4 + base0 + lane, bt + lane);
    async_cp_b128(w4 + base0 + lane, bw + lane);
  }
  for (long long base = base0; base < total4; base += stride) {
    const long long nbase = base + stride;
    const bool nextFull = (nbase < total4) && (nbase + 32 <= total4);
    const int nxt = cur ^ 1;
    if (nextFull) {
      async_cp_b128(p4 + nbase + lane, bp + nxt * 32 + lane);
      async_cp_b128(t4 + nbase + lane, bt + nxt * 32 + lane);
      async_cp_b128(w4 + nbase + lane, bw + nxt * 32 + lane);
    }
    float4 p = {0.f, 0.f, 0.f, 0.f}, t = {0.f, 0.f, 0.f, 0.f},
           w = {0.f, 0.f, 0.f, 0.f};
    bool act;
    if (curFull) {
      if (nextFull) wait_async3(); else wait_async0();
      p = bp[cur * 32 + lane];
      t = bt[cur * 32 + lane];
      w = bw[cur * 32 + lane];
      act = true;
    } else {  // partial wave-tile (at most once): direct guarded loads
      const long long i = base + lane;
      act = i < total4;
      if (act) { p = p4[i]; t = t4[i]; w = w4[i]; }
    }
    fn(p, t, w, act);
    cur = nxt;
    curFull = nextFull;
  }
#else
  (void)bp; (void)bt; (void)bw;
  for (long long base = base0; base < total4; base += stride) {
    const long long i = base + lane;
    const bool act = i < total4;
    float4 p = {0.f, 0.f, 0.f, 0.f}, t = {0.f, 0.f, 0.f, 0.f},
           w = {0.f, 0.f, 0.f, 0.f};
    if (act) {
      p = p4[i]; t = t4[i]; w = w4[i];
      __builtin_prefetch((const void*)(p4 + i + stride), 0, 0);
      __builtin_prefetch((const void*)(t4 + i + stride), 0, 0);
      __builtin_prefetch((const void*)(w4 + i + stride), 0, 0);
    }
    fn(p, t, w, act);
  }
#endif
}

// ---------------------------------------------------------------------------
// kernel 0: zero workspace accumulators (graph replay => must re-zero)
// ---------------------------------------------------------------------------
__global__ void ghm_zero(unsigned* __restrict__ counts, float* __restrict__ acc) {
  int t = threadIdx.x;
  if (t < GHM_BINS) counts[t] = 0u;
  if (t == GHM_BINS) *acc = 0.0f;
}

// ---------------------------------------------------------------------------
// kernel 1: 10-bin histogram of valid elements (wave32 ballots, lane b counts
// bin b; flushed via LDS ds_add_u32, then 10 global atomics per block)
// ---------------------------------------------------------------------------
__global__ __launch_bounds__(THREADS) void ghm_hist(
    const float* __restrict__ pred, const float* __restrict__ tgt,
    const float* __restrict__ lw, unsigned* __restrict__ g_counts, long long n) {
  __shared__ unsigned s_bins[GHM_BINS];
  const int tid = threadIdx.x;
  const int lane = tid & 31;
  const int wv = tid >> 5;
  if (tid < GHM_BINS) s_bins[tid] = 0u;
  __syncthreads();

#if HAS_ASYNC_LDS
  __shared__ float4 s_stage[NWAVES][3][2][32];
  float4* bp = &s_stage[wv][0][0][0];
  float4* bt = &s_stage[wv][1][0][0];
  float4* bw = &s_stage[wv][2][0][0];
#else
  float4 *bp = nullptr, *bt = nullptr, *bw = nullptr;
#endif

  const long long total4 = n >> 2;
  unsigned my_cnt = 0;  // lane b (<10) accumulates bin b

  stream_tiles((const float4*)pred, (const float4*)tgt, (const float4*)lw,
               total4, bp, bt, bw,
               [&](float4 p, float4 t, float4 w, bool act) {
                 const int b0 = act ? ghm_bin(p.x, t.x, w.x) : -1;
                 const int b1 = act ? ghm_bin(p.y, t.y, w.y) : -1;
                 const int b2 = act ? ghm_bin(p.z, t.z, w.z) : -1;
                 const int b3 = act ? ghm_bin(p.w, t.w, w.w) : -1;
#pragma unroll
                 for (int bb = 0; bb < GHM_BINS; ++bb) {
                   unsigned c = __popc(ballot32(b0 == bb)) +
                                __popc(ballot32(b1 == bb)) +
                                __popc(ballot32(b2 == bb)) +
                                __popc(ballot32(b3 == bb));
                   if (lane == bb) my_cnt += c;
                 }
               });

  // tail (n % 4): wave 0 of block 0 only (full wave -> ballots safe)
  const int rem = (int)(n - (total4 << 2));
  if (rem > 0 && blockIdx.x == 0 && tid < 32) {
    const long long base = total4 << 2;
    const int b = (lane < rem)
                      ? ghm_bin(pred[base + lane], tgt[base + lane], lw[base + lane])
                      : -1;
#pragma unroll
    for (int bb = 0; bb < GHM_BINS; ++bb) {
      unsigned c = __popc(ballot32(b == bb));
      if (lane == bb) my_cnt += c;
    }
  }

  if (lane < GHM_BINS && my_cnt) atomicAdd(&s_bins[lane], my_cnt);  // ds_add_u32
  __syncthreads();
  if (tid < GHM_BINS) {
    unsigned c = s_bins[tid];
    if (c) atomicAdd(&g_counts[tid], c);
  }
}

// ---------------------------------------------------------------------------
// kernel 2: counts -> k[b] = 1/(n_bins * max(c_b,1)) (0 if no valid elements);
// tot cancels: sum(w*bce)/tot == sum_valid bce/(n*max(c,1)).  Zeroes acc.
// ---------------------------------------------------------------------------
__global__ void ghm_finalize(const unsigned* __restrict__ counts,
                             float* __restrict__ kvec, float* __restrict__ acc) {
  if (threadIdx.x == 0) {
    int nb = 0;
#pragma unroll
    for (int b = 0; b < GHM_BINS; ++b) nb += (counts[b] > 0u) ? 1 : 0;
    const float inv_n = (nb > 0) ? (1.0f / (float)nb) : 0.0f;
#pragma unroll
    for (int b = 0; b < GHM_BINS; ++b) {
      unsigned c = counts[b];
      kvec[b] = inv_n / (float)(c > 0u ? c : 1u);
    }
    *acc = 0.0f;
  }
}

// ---------------------------------------------------------------------------
// kernel 3: weighted-BCE reduction; WMMA wave sum -> 8 partials -> 1 atomic
// ---------------------------------------------------------------------------
__device__ __forceinline__ float ghm_contrib(float p, float t, float w,
                                             const float* __restrict__ s_k) {
  if (!(w > 0.0f)) return 0.0f;
  float s = 1.0f / (1.0f + expf(-p));
  int b = (int)(fabsf(s - t) * (float)GHM_BINS);
  b = b < 0 ? 0 : (b > GHM_BINS - 1 ? GHM_BINS - 1 : b);
  return s_k[b] * bce_logits(p, t);
}

__global__ __launch_bounds__(THREADS) void ghm_loss(
    const float* __restrict__ pred, const float* __restrict__ tgt,
    const float* __restrict__ lw, const float* __restrict__ kvec,
    float* __restrict__ acc, long long n) {
  __shared__ float s_k[GHM_BINS];
  __shared__ float s_part[NWAVES];
  const int tid = threadIdx.x;
  const int lane = tid & 31;
  const int wv = tid >> 5;
  if (tid < GHM_BINS) s_k[tid] = kvec[tid];
  __syncthreads();

#if HAS_ASYNC_LDS
  __shared__ float4 s_stage[NWAVES][3][2][32];
  float4* bp = &s_stage[wv][0][0][0];
  float4* bt = &s_stage[wv][1][0][0];
  float4* bw = &s_stage[wv][2][0][0];
#else
  float4 *bp = nullptr, *bt = nullptr, *bw = nullptr;
#endif

  const long long total4 = n >> 2;
  float a = 0.0f;

  stream_tiles((const float4*)pred, (const float4*)tgt, (const float4*)lw,
               total4, bp, bt, bw,
               [&](float4 p, float4 t, float4 w, bool act) {
                 if (act) {
                   a += ghm_contrib(p.x, t.x, w.x, s_k);
                   a += ghm_contrib(p.y, t.y, w.y, s_k);
                   a += ghm_contrib(p.z, t.z, w.z, s_k);
                   a += ghm_contrib(p.w, t.w, w.w, s_k);
                 }
               });

  // tail (n % 4)
  const int rem = (int)(n - (total4 << 2));
  if (rem > 0 && blockIdx.x == 0 && tid < rem) {
    const long long base = total4 << 2;
    a += ghm_contrib(pred[base + tid], tgt[base + tid], lw[base + tid], s_k);
  }

  // full-wave reduction (EXEC all-1s here): v_wmma_f32_16x16x4_f32 x5
  a = wave_sum32(a);
  if (lane == 0) s_part[wv] = a;
  __syncthreads();
  if (tid == 0) {
    float s = 0.0f;
#pragma unroll
    for (int i = 0; i < NWAVES; ++i) s += s_part[i];
    atomicAdd(acc, s);
  }
}

// kernel 4: publish scalar (LOSS_WEIGHT == 1)
__global__ void ghm_write(const float* __restrict__ acc, float* __restrict__ out) {
  if (threadIdx.x == 0) out[0] = acc[0];
}

// ---------------------------------------------------------------------------
extern "C" void kernel_launch(void* const* d_in, const int* in_sizes, int n_in,
                              void* d_out, int out_size, void* d_ws, size_t ws_size,
                              hipStream_t stream) {
  const float* pred = (const float*)d_in[0];
  const float* tgt  = (const float*)d_in[1];
  const float* lw   = (const float*)d_in[2];
  const long long n = (long long)in_sizes[0];

  unsigned char* ws = (unsigned char*)d_ws;
  unsigned* counts = (unsigned*)ws;        // 10 * 4 B
  float*    kvec   = (float*)(ws + 64);    // 10 * 4 B
  float*    acc    = (float*)(ws + 128);   // 4 B

  const long long total4 = n >> 2;
  long long want = (total4 + THREADS - 1) / THREADS;
  if (want < 1) want = 1;
  const int blocks = (int)(want > 16384 ? 16384 : want);

  ghm_zero<<<1, 32, 0, stream>>>(counts, acc);
  ghm_hist<<<blocks, THREADS, 0, stream>>>(pred, tgt, lw, counts, n);
  ghm_finalize<<<1, 32, 0, stream>>>(counts, kvec, acc);
  ghm_loss<<<blocks, THREADS, 0, stream>>>(pred, tgt, lw, kvec, acc, n);
  ghm_write<<<1, 32, 0, stream>>>(acc, (float*)d_out);
}